// GraphSAGE_21955872817707
// MI455X (gfx1250) — compile-verified
//
#include <hip/hip_runtime.h>
#include <hip/hip_bf16.h>

typedef __attribute__((ext_vector_type(2))) float v2f;
typedef __attribute__((ext_vector_type(8))) float v8f;

#define BATCH  4
#define CIN    192
#define NNODES 8192
#define KNBR   16
#define COUT   192

// ---------------------------------------------------------------------------
// Kernel 1: y[b] = relu(W1 (192x192) @ x[b] (192x8192) + b1)
// Wave tiling: each wave computes a 16(M) x 64(N) tile via 4 accumulators,
// K-loop in steps of 4 using V_WMMA_F32_16X16X4_F32.
// y layout: (B, C, N) row-major, written into d_out (scratch until gemm2).
// ---------------------------------------------------------------------------
__global__ __launch_bounds__(256) void gemm1_kernel(const float* __restrict__ x,
                                                    const float* __restrict__ w1,
                                                    const float* __restrict__ b1,
                                                    float* __restrict__ y)
{
    const int wid  = threadIdx.x >> 5;
    const int lane = threadIdx.x & 31;
    const int half = lane >> 4;     // 0: K pair {0,1}, 1: K pair {2,3}
    const int l    = lane & 15;

    const int wg = blockIdx.x * 8 + wid;      // global wave id
    const int nt = wg & 127;                  // 128 n-tiles of width 64
    const int t  = wg >> 7;
    const int mo = t % (CIN / 16);            // 12 m-tiles
    const int b  = t / (CIN / 16);

    const int n0 = nt * 64;
    const int m0 = mo * 16;

    const float* xb = x + (size_t)b * CIN * NNODES;
    const float* ap = w1 + (size_t)(m0 + l) * CIN + 2 * half;
    const float* bp = xb + (size_t)(2 * half) * NNODES + n0 + l;

    v8f acc0 = {}, acc1 = {}, acc2 = {}, acc3 = {};

#pragma unroll 4
    for (int c0 = 0; c0 < CIN; c0 += 4) {
        v2f a = *(const v2f*)(ap + c0);                 // A 16x4 fragment
        const float* bpc = bp + (size_t)c0 * NNODES;
        v2f b0; b0.x = bpc[0];       b0.y = bpc[NNODES];
        v2f b1v; b1v.x = bpc[16];    b1v.y = bpc[16 + NNODES];
        v2f b2v; b2v.x = bpc[32];    b2v.y = bpc[32 + NNODES];
        v2f b3v; b3v.x = bpc[48];    b3v.y = bpc[48 + NNODES];
        acc0 = __builtin_amdgcn_wmma_f32_16x16x4_f32(false, a, false, b0,  (short)0, acc0, false, false);
        acc1 = __builtin_amdgcn_wmma_f32_16x16x4_f32(false, a, false, b1v, (short)0, acc1, false, false);
        acc2 = __builtin_amdgcn_wmma_f32_16x16x4_f32(false, a, false, b2v, (short)0, acc2, false, false);
        acc3 = __builtin_amdgcn_wmma_f32_16x16x4_f32(false, a, false, b3v, (short)0, acc3, false, false);
    }

#pragma unroll
    for (int r = 0; r < 8; ++r) {
        const int m = m0 + r + 8 * half;                // D-layout: M = r + 8*(lane>=16)
        const float bias = b1[m];
        float* yrow = y + ((size_t)b * CIN + m) * NNODES + n0 + l;
        float v0 = acc0[r] + bias; yrow[0]  = v0 > 0.f ? v0 : 0.f;
        float v1 = acc1[r] + bias; yrow[16] = v1 > 0.f ? v1 : 0.f;
        float v2 = acc2[r] + bias; yrow[32] = v2 > 0.f ? v2 : 0.f;
        float v3 = acc3[r] + bias; yrow[48] = v3 > 0.f ? v3 : 0.f;
    }
}

// ---------------------------------------------------------------------------
// Kernel 2: h[b,o,n] = max_k y[b,o,idx[b,n,k]]
// Block = 32 nodes x all 192 channels (8 waves stride the channels).
// Each lane caches its node's 16 neighbor indices in registers (int4 loads),
// gathers from the L2-resident y rows, writes h coalesced.
// ---------------------------------------------------------------------------
__global__ __launch_bounds__(256) void gathermax_kernel(const float* __restrict__ y,
                                                        const int* __restrict__ idx,
                                                        float* __restrict__ h)
{
    const int blk  = blockIdx.x;          // BATCH * (NNODES/32)
    const int b    = blk >> 8;            // NNODES/32 == 256
    const int n0   = (blk & 255) * 32;
    const int wid  = threadIdx.x >> 5;
    const int lane = threadIdx.x & 31;

    const int4* ip = (const int4*)(idx + ((size_t)b * NNODES + n0 + lane) * KNBR);
    const int4 j0 = ip[0], j1 = ip[1], j2 = ip[2], j3 = ip[3];

    for (int o = wid; o < CIN; o += 8) {
        const float* yr = y + ((size_t)b * CIN + o) * NNODES;
        float m0 = fmaxf(fmaxf(yr[j0.x], yr[j0.y]), fmaxf(yr[j0.z], yr[j0.w]));
        float m1 = fmaxf(fmaxf(yr[j1.x], yr[j1.y]), fmaxf(yr[j1.z], yr[j1.w]));
        float m2 = fmaxf(fmaxf(yr[j2.x], yr[j2.y]), fmaxf(yr[j2.z], yr[j2.w]));
        float m3 = fmaxf(fmaxf(yr[j3.x], yr[j3.y]), fmaxf(yr[j3.z], yr[j3.w]));
        h[((size_t)b * CIN + o) * NNODES + n0 + lane] = fmaxf(fmaxf(m0, m1), fmaxf(m2, m3));
    }
}

// ---------------------------------------------------------------------------
// Kernel 3: out = relu(W2 (192x384) @ [x; h] + b2)
// Same WMMA tiling as gemm1; K dimension = 384, B operand switches from x to
// h at c0 == CIN (uniform branch, never straddles a K-step of 4).
// ---------------------------------------------------------------------------
__global__ __launch_bounds__(256) void gemm2_kernel(const float* __restrict__ x,
                                                    const float* __restrict__ h,
                                                    const float* __restrict__ w2,
                                                    const float* __restrict__ b2,
                                                    float* __restrict__ out)
{
    const int wid  = threadIdx.x >> 5;
    const int lane = threadIdx.x & 31;
    const int half = lane >> 4;
    const int l    = lane & 15;

    const int wg = blockIdx.x * 8 + wid;
    const int nt = wg & 127;
    const int t  = wg >> 7;
    const int mo = t % (COUT / 16);
    const int b  = t / (COUT / 16);

    const int n0 = nt * 64;
    const int m0 = mo * 16;

    const float* xb = x + (size_t)b * CIN * NNODES;
    const float* hb = h + (size_t)b * CIN * NNODES;
    const float* ap = w2 + (size_t)(m0 + l) * (2 * CIN) + 2 * half;

    v8f acc0 = {}, acc1 = {}, acc2 = {}, acc3 = {};

#pragma unroll 4
    for (int c0 = 0; c0 < 2 * CIN; c0 += 4) {
        v2f a = *(const v2f*)(ap + c0);
        const float* src = (c0 < CIN) ? (xb + (size_t)c0 * NNODES)
                                      : (hb + (size_t)(c0 - CIN) * NNODES);
        const float* bpc = src + (size_t)(2 * half) * NNODES + n0 + l;
        v2f b0; b0.x = bpc[0];       b0.y = bpc[NNODES];
        v2f b1v; b1v.x = bpc[16];    b1v.y = bpc[16 + NNODES];
        v2f b2v; b2v.x = bpc[32];    b2v.y = bpc[32 + NNODES];
        v2f b3v; b3v.x = bpc[48];    b3v.y = bpc[48 + NNODES];
        acc0 = __builtin_amdgcn_wmma_f32_16x16x4_f32(false, a, false, b0,  (short)0, acc0, false, false);
        acc1 = __builtin_amdgcn_wmma_f32_16x16x4_f32(false, a, false, b1v, (short)0, acc1, false, false);
        acc2 = __builtin_amdgcn_wmma_f32_16x16x4_f32(false, a, false, b2v, (short)0, acc2, false, false);
        acc3 = __builtin_amdgcn_wmma_f32_16x16x4_f32(false, a, false, b3v, (short)0, acc3, false, false);
    }

#pragma unroll
    for (int r = 0; r < 8; ++r) {
        const int m = m0 + r + 8 * half;
        const float bias = b2[m];
        float* orow = out + ((size_t)b * COUT + m) * NNODES + n0 + l;
        float v0 = acc0[r] + bias; orow[0]  = v0 > 0.f ? v0 : 0.f;
        float v1 = acc1[r] + bias; orow[16] = v1 > 0.f ? v1 : 0.f;
        float v2 = acc2[r] + bias; orow[32] = v2 > 0.f ? v2 : 0.f;
        float v3 = acc3[r] + bias; orow[48] = v3 > 0.f ? v3 : 0.f;
    }
}

extern "C" void kernel_launch(void* const* d_in, const int* in_sizes, int n_in,
                              void* d_out, int out_size, void* d_ws, size_t ws_size,
                              hipStream_t stream) {
    const float* x    = (const float*)d_in[0];   // (B, C, N, 1)
    const int*   edge = (const int*)d_in[1];     // (2, B, N, K); idx = edge_index[0]
    const float* w1   = (const float*)d_in[2];   // (C, C)
    const float* b1   = (const float*)d_in[3];   // (C,)
    const float* w2   = (const float*)d_in[4];   // (OUT, 2C)
    const float* b2   = (const float*)d_in[5];   // (OUT,)
    float* out = (float*)d_out;                  // (B, OUT, N, 1)

    // y (B,C,N) lives in d_out: it is dead before gemm2 overwrites d_out.
    float* y = (float*)d_out;
    // h (B,C,N) in workspace: 4*192*8192*4 = 25.2 MB
    float* h = (float*)d_ws;
    const int* idx = edge;                       // first B*N*K ints = edge_index[0]

    const int gemm_blocks = (BATCH * (CIN / 16) * (NNODES / 64)) / 8;   // 768
    const int gat_blocks  = BATCH * (NNODES / 32);                      // 1024

    gemm1_kernel<<<gemm_blocks, 256, 0, stream>>>(x, w1, b1, y);
    gathermax_kernel<<<gat_blocks, 256, 0, stream>>>(y, idx, h);
    gemm2_kernel<<<gemm_blocks, 256, 0, stream>>>(x, h, w2, b2, out);
}